// FlowEmbedder_33552284516523
// MI455X (gfx1250) — compile-verified
//
#include <hip/hip_runtime.h>

// ---------------------------------------------------------------------------
// FlowEmbedder cost matrix on gfx1250 (MI455X).
//
// dot(i,j,t) is cubic in t:  dot = c0 + c1 t + c2 t^2 + c3 t^3
// For 16 pairs x 16 t-values this is a (16x4)x(4x16) product ->
// one V_WMMA_F32_16X16X4_F32 per 16 output elements.
//   A (16x4): Vandermonde in t (constant):
//       lanes 0-15 : VGPR0=1,    VGPR1=t      (t = lane)
//       lanes16-31 : VGPR0=t^2,  VGPR1=t^3    (t = lane-16)
//   B (4x16): per-pair coefficients (column-major):
//       lanes 0-15 : VGPR0=c0,   VGPR1=c1     (pair = lane)
//       lanes16-31 : VGPR0=c2,   VGPR1=c3     (pair = lane-16)
//   D (16x16): lane L, VGPR r => pair = L&15, t = r + 8*(L>>4)
// Each lane does 8x v_exp_f32 + mask + local sum, then a ds_swizzle
// SWAPX16 combines the two t-halves; lanes 0-15 store 16 contiguous floats.
// ---------------------------------------------------------------------------

typedef __attribute__((ext_vector_type(2))) float v2f;
typedef __attribute__((ext_vector_type(8))) float v8f;
typedef __attribute__((ext_vector_type(4))) int   v4i;

#define NPTS        2048
#define STEPSZ      0.1f
#define ROWS_PER_WAVE 4
#define WAVES_PER_BLK 8
#define TILE_J      16
#define TILE_I      (ROWS_PER_WAVE * WAVES_PER_BLK)   // 32

#define HAS_ASYNC_LDS ( __has_builtin(__builtin_amdgcn_global_load_async_to_lds_b128) && \
                        __has_builtin(__builtin_amdgcn_s_wait_asynccnt) )

// Exact-typed pointer casts for the async-to-LDS builtin (param types taken
// from the compiler diagnostic: int4 in AS1 / AS3).
#define ASYNC_GPTR(p) ((__attribute__((address_space(1))) v4i*)(p))
#define ASYNC_LPTR(p) ((__attribute__((address_space(3))) v4i*)(p))

__global__ __launch_bounds__(256) void flow_cost_kernel(
    const float* __restrict__ pts,   // [NPTS, 2]
    const float* __restrict__ prm,   // [3, 2]
    float* __restrict__ out)         // [NPTS, NPTS]
{
    __shared__ float lds_j[TILE_J * 2];   // 128 B, 16-B aligned chunks
    __shared__ float lds_i[TILE_I * 2];   // 256 B

    const int tid  = threadIdx.x;
    const int lane = tid & 31;
    const int wave = tid >> 5;
    const int jb   = blockIdx.x * TILE_J;
    const int ib   = blockIdx.y * TILE_I;

    // ---- Stage this block's point tiles into LDS (async DMA path) ----------
    if (wave == 0) {
#if HAS_ASYNC_LDS
        if (lane < 24) {                       // 24 x b128 = 384 bytes total
            const int isJ = (lane < 8);
            const int l   = isJ ? lane : (lane - 8);
            const float* gsrc = isJ ? (pts + (size_t)jb * 2 + l * 4)
                                    : (pts + (size_t)ib * 2 + l * 4);
            float* ldst = isJ ? (lds_j + l * 4) : (lds_i + l * 4);
            __builtin_amdgcn_global_load_async_to_lds_b128(
                ASYNC_GPTR(gsrc), ASYNC_LPTR(ldst), 0, 0);
        }
        __builtin_amdgcn_s_wait_asynccnt(0);   // issuing wave drains before barrier
#else
        if (lane < 24) {
            const int isJ = (lane < 8);
            const int l   = isJ ? lane : (lane - 8);
            const float4 v = isJ ? ((const float4*)(pts + (size_t)jb * 2))[l]
                                 : ((const float4*)(pts + (size_t)ib * 2))[l];
            if (isJ) ((float4*)lds_j)[l] = v; else ((float4*)lds_i)[l] = v;
        }
#endif
    }
    __syncthreads();

    // ---- Uniform flow-field weights (w[k] = comb(3,k) * params[k]) --------
    const float w00 = prm[0],        w01 = prm[1];
    const float w10 = 3.0f * prm[2], w11 = 3.0f * prm[3];
    const float w20 = 3.0f * prm[4], w21 = 3.0f * prm[5];

    const int   jl   = lane & 15;          // pair-in-tile AND t-index for A
    const int   half = lane >> 4;
    const float xj   = lds_j[jl * 2 + 0];
    const float yj   = lds_j[jl * 2 + 1];

    // Constant A matrix (Vandermonde in t), per 32-bit A 16x4 VGPR layout.
    const float tA = (float)jl;
    v2f A;
    A.x = half ? (tA * tA)      : 1.0f;
    A.y = half ? (tA * tA * tA) : tA;

    const int j = jb + jl;

    for (int r = 0; r < ROWS_PER_WAVE; ++r) {
        const int   i  = ib + wave * ROWS_PER_WAVE + r;
        const int   il = wave * ROWS_PER_WAVE + r;
        const float xi = lds_i[il * 2 + 0];
        const float yi = lds_i[il * 2 + 1];

        const float dx   = xj - xi;
        const float dy   = yj - yi;
        const float dist = __builtin_sqrtf(dx * dx + dy * dy);
        const float safe = (dist > 0.0f) ? dist : 1.0f;
        const float s_in = STEPSZ / safe;          // step_size / ||diff||
        const float u    = dx * s_in;              // per-t displacement
        const float v    = dy * s_in;

        const float a0 = w00 * dx + w01 * dy;      // comb-weighted param . diff
        const float a1 = w10 * dx + w11 * dy;
        const float a2 = w20 * dx + w21 * dy;

        // dot(t) = c0 + c1 t + c2 t^2 + c3 t^3  (expansion of the cubic flow)
        const float c0 = xi * (a0 * xi * xi + (a1 * xi + a2 * yi) * yi);
        const float c1 = a0 * 3.0f * xi * xi * u
                       + a1 * (2.0f * xi * u * yi + xi * xi * v)
                       + a2 * (u * yi * yi + 2.0f * xi * yi * v);
        const float c2 = a0 * 3.0f * xi * u * u
                       + a1 * (u * u * yi + 2.0f * xi * u * v)
                       + a2 * (2.0f * u * yi * v + xi * v * v);
        const float c3 = u * (a0 * u * u + a1 * u * v + a2 * v * v);

        v2f B;                                     // column-major B 4x16
        B.x = half ? c2 : c0;
        B.y = half ? c3 : c1;

        v8f C = {};
        // D[t, pair] = sum_k t^k * c_k(pair)   (one WMMA per 16 outputs)
        v8f D = __builtin_amdgcn_wmma_f32_16x16x4_f32(
            false, A, false, B, (short)0, C, false, false);

        const float nsteps = __builtin_floorf(dist * 10.0f);
        const float tbase  = (float)(half * 8);
        float acc = 0.0f;
#pragma unroll
        for (int m = 0; m < 8; ++m) {
            const float t   = tbase + (float)m;
            // 1/cost_normalizer = (1 + (FLOW-1) e^{-dot}) / FLOW
            const float val = 0.2f + 0.8f * __expf(-D[m]);
            acc += (t <= nsteps) ? val : 0.0f;
        }

        // Combine lane L with lane L^16 (other half of the t range, same pair)
        const int other = __builtin_amdgcn_ds_swizzle(__float_as_int(acc), 0x401F);
        acc = (acc + __int_as_float(other)) * STEPSZ;

        if (half == 0) out[(size_t)i * NPTS + j] = acc;
    }
}

extern "C" void kernel_launch(void* const* d_in, const int* in_sizes, int n_in,
                              void* d_out, int out_size, void* d_ws, size_t ws_size,
                              hipStream_t stream) {
    const float* pts = (const float*)d_in[0];   // embedded_points [2048,2]
    const float* prm = (const float*)d_in[1];   // flow_field_parameters [3,2]
    float* out = (float*)d_out;                 // [2048,2048]

    dim3 grid(NPTS / TILE_J, NPTS / TILE_I);    // 128 x 64
    dim3 block(256);                            // 8 wave32
    flow_cost_kernel<<<grid, block, 0, stream>>>(pts, prm, out);
}